// GraphConvLayer_35691178230263
// MI455X (gfx1250) — compile-verified
//
#include <hip/hip_runtime.h>

// GraphConv scatter-gather for MI455X (gfx1250, wave32).
//   y = segment_sum(Param_W[widx] * x[src], dst) + Param_b[node_label]
//
// Memory-bound (AI ~0.17 FLOP/B):
//  * src/dst/widx streamed once (153.6 MB total) -> b128 loads, th:TH_LOAD_NT
//    so the one-shot stream does not evict the hot tables from L2/WGP$.
//  * Param_W (4 MB) + x (800 KB) gathers stay resident in the 192 MB L2.
//  * y (800 KB) scatter-adds via no-return global_atomic_add_f32 (STOREcnt),
//    resolved by the L2 near-memory atomic units.
//  * 8 edges/thread in two independent 4-edge groups; the scheduler can
//    interleave them freely (partial loadcnt waits, deeper pipelining).

#define THREADS 256
#define EPT 8  // edges per thread

typedef int v4i __attribute__((ext_vector_type(4)));

__global__ __launch_bounds__(THREADS) void gc_edge_scatter(
    const float* __restrict__ x,
    const float* __restrict__ W,
    const int*   __restrict__ src,
    const int*   __restrict__ dst,
    const int*   __restrict__ widx,
    float*       __restrict__ y,
    int E) {
  int t    = blockIdx.x * THREADS + threadIdx.x;
  int base = t * EPT;

  if (base + EPT <= E) {
    // ---- group 0: edges [base, base+4) ----
    v4i s0 = __builtin_nontemporal_load((const v4i*)(src  + base));
    v4i d0 = __builtin_nontemporal_load((const v4i*)(dst  + base));
    v4i w0 = __builtin_nontemporal_load((const v4i*)(widx + base));

    float m0 = W[w0.x] * x[s0.x];
    float m1 = W[w0.y] * x[s0.y];
    float m2 = W[w0.z] * x[s0.z];
    float m3 = W[w0.w] * x[s0.w];

    unsafeAtomicAdd(y + d0.x, m0);
    unsafeAtomicAdd(y + d0.y, m1);
    unsafeAtomicAdd(y + d0.z, m2);
    unsafeAtomicAdd(y + d0.w, m3);

    // ---- group 1: edges [base+4, base+8) ----
    v4i s1 = __builtin_nontemporal_load((const v4i*)(src  + base + 4));
    v4i d1 = __builtin_nontemporal_load((const v4i*)(dst  + base + 4));
    v4i w1 = __builtin_nontemporal_load((const v4i*)(widx + base + 4));

    float m4 = W[w1.x] * x[s1.x];
    float m5 = W[w1.y] * x[s1.y];
    float m6 = W[w1.z] * x[s1.z];
    float m7 = W[w1.w] * x[s1.w];

    unsafeAtomicAdd(y + d1.x, m4);
    unsafeAtomicAdd(y + d1.y, m5);
    unsafeAtomicAdd(y + d1.z, m6);
    unsafeAtomicAdd(y + d1.w, m7);
  } else if (base < E) {
    // Tail (E not a multiple of EPT) — scalar cleanup.
    for (int e = base; e < E; ++e) {
      unsafeAtomicAdd(y + dst[e], W[widx[e]] * x[src[e]]);
    }
  }
}

__global__ __launch_bounds__(THREADS) void gc_bias_init(
    const float* __restrict__ Pb,
    const int*   __restrict__ label,
    float*       __restrict__ y,
    int n) {
  int i = blockIdx.x * THREADS + threadIdx.x;
  if (i < n) {
    y[i] = Pb[label[i]];
  }
}

extern "C" void kernel_launch(void* const* d_in, const int* in_sizes, int n_in,
                              void* d_out, int out_size, void* d_ws, size_t ws_size,
                              hipStream_t stream) {
  // setup_inputs() order:
  //   0: x          float32 [N_NODES]
  //   1: Param_W    float32 [WEIGHT_NUM]
  //   2: Param_b    float32 [BIAS_NUM]
  //   3: src        int32   [N_EDGES]
  //   4: dst        int32   [N_EDGES]
  //   5: weight_idx int32   [N_EDGES]
  //   6: node_label int32   [N_NODES]
  const float* x    = (const float*)d_in[0];
  const float* W    = (const float*)d_in[1];
  const float* Pb   = (const float*)d_in[2];
  const int*   src  = (const int*)d_in[3];
  const int*   dst  = (const int*)d_in[4];
  const int*   widx = (const int*)d_in[5];
  const int*   lab  = (const int*)d_in[6];
  float*       y    = (float*)d_out;

  const int N = in_sizes[0];
  const int E = in_sizes[3];

  // 1) y = Param_b[node_label]  (also initializes the poisoned output).
  int bias_blocks = (N + THREADS - 1) / THREADS;
  gc_bias_init<<<bias_blocks, THREADS, 0, stream>>>(Pb, lab, y, N);

  // 2) Scatter-add edge messages (stream-ordered after bias init).
  int nthreads    = (E + EPT - 1) / EPT;
  int edge_blocks = (nthreads + THREADS - 1) / THREADS;
  gc_edge_scatter<<<edge_blocks, THREADS, 0, stream>>>(x, W, src, dst, widx, y, E);
}